// MultiGaussianReadoutLayer_29326036697222
// MI455X (gfx1250) — compile-verified
//
#include <hip/hip_runtime.h>

// Problem constants (from reference)
#define B_   32
#define H_   24
#define N_   500
#define FIN  64
#define FOUT 4
#define M_   16
#define O_   (H_ * N_)   // 12000 instances

typedef float v2f __attribute__((ext_vector_type(2)));
typedef float v8f __attribute__((ext_vector_type(8)));

#define XS_STRIDE  68   // 32x64 x-tile, padded: bank = (68*row + k) % 64 conflict-free for 16 rows
#define SIG_STRIDE 17   // 32x16 sigma, padded: distinct banks across 32 rows
#define MU_STRIDE  5

__global__ __launch_bounds__(256)
void mgr_fused(const float* __restrict__ x,        // [B, O, FIN]
               const float* __restrict__ mu_w,     // [O, FIN, FOUT]
               const float* __restrict__ mu_b,     // [O, FOUT]
               const float* __restrict__ sigma_w,  // [O, FIN, 16]
               const float* __restrict__ sigma_b,  // [O, 16]
               const float* __restrict__ eps,      // [M, B, O, FOUT]
               float* __restrict__ out) {          // [M, B, O, FOUT]
  __shared__ float xs [B_ * XS_STRIDE];   // x tile for this o
  __shared__ float sw [FIN * 16];         // sigma_w[o], row-major [k][n]
  __shared__ float mwl[FIN * FOUT];       // mu_w[o]
  __shared__ float sig[B_ * SIG_STRIDE];  // sigma result [b][16]
  __shared__ float mus[B_ * MU_STRIDE];   // mu result [b][4]

  const int o    = blockIdx.x;            // instance index (h*N + n)
  const int tid  = threadIdx.x;
  const int lane = tid & 31;              // wave32
  const int wid  = tid >> 5;              // wave id, uniform per wave

  // ---- eps prefetch: independent of GEMM, overlaps with compute ----
  const int m0 = tid >> 5;                // 0..7
  const int b0 = tid & 31;                // 0..31
  const float4* epsv = (const float4*)eps;
  float4 e0 = epsv[((size_t)m0       * B_ + b0) * O_ + o];
  float4 e1 = epsv[((size_t)(m0 + 8) * B_ + b0) * O_ + o];

  // ---- coalesced staging into LDS ----
  {
    const float4* xv = (const float4*)x;  // row stride O_*FIN/4 float4
#pragma unroll
    for (int i = 0; i < 2; ++i) {
      int f4  = tid + 256 * i;            // 0..511 float4 of the 32x64 tile
      int row = f4 >> 4;                  // b
      int c4  = f4 & 15;                  // float4 column
      float4 v = xv[(size_t)row * (O_ * (FIN / 4)) + (size_t)o * (FIN / 4) + c4];
      *(float4*)&xs[row * XS_STRIDE + c4 * 4] = v;   // 272*row + 16*c4 bytes: 16B aligned
    }
    const float4* swv = (const float4*)(sigma_w + (size_t)o * (FIN * 16));
    ((float4*)sw)[tid] = swv[tid];        // 256 float4 = 64x16 tile
    if (tid < 64) {
      const float4* mwv = (const float4*)(mu_w + (size_t)o * (FIN * FOUT));
      ((float4*)mwl)[tid] = mwv[tid];     // 64 float4 = 64x4 tile
    }
  }
  __syncthreads();

  // ---- compute stage ----
  if (wid < 2) {
    // sigma GEMM tile: rows b = wid*16 .. wid*16+15, cols 0..15, K = 64
    // via V_WMMA_F32_16X16X4_F32, C preloaded with sigma_b broadcast.
    const int col  = lane & 15;           // n (and A-row m) per ISA layout
    const int hi   = lane >> 4;           // selects K pair {0,1} vs {2,3}
    const int arow = wid * 16 + col;      // b row this lane feeds for A

    float sb = sigma_b[(size_t)o * 16 + col];
    v8f c = {sb, sb, sb, sb, sb, sb, sb, sb};

#pragma unroll
    for (int k0 = 0; k0 < FIN; k0 += 4) {
      const int ka = k0 + 2 * hi;
      v2f a, b;
      a.x = xs[arow * XS_STRIDE + ka];        // A[m][ka], A[m][ka+1]
      a.y = xs[arow * XS_STRIDE + ka + 1];
      b.x = sw[ka * 16 + col];                // B[ka][n], B[ka+1][n]
      b.y = sw[(ka + 1) * 16 + col];
      c = __builtin_amdgcn_wmma_f32_16x16x4_f32(
              /*neg_a=*/false, a, /*neg_b=*/false, b,
              /*c_mod=*/(short)0, c, /*reuse_a=*/false, /*reuse_b=*/false);
    }
    // C layout: VGPR r -> M = r (lanes 0-15) / r+8 (lanes 16-31), N = lane&15
#pragma unroll
    for (int r = 0; r < 8; ++r) {
      int bb = wid * 16 + hi * 8 + r;
      sig[bb * SIG_STRIDE + col] = c[r];
    }
  } else if (wid < 6) {
    // mu GEMM: 32x4, K=64 — tiny, plain FMA from LDS
    int q  = tid - 64;                    // 0..127
    int bb = q & 31;
    int g  = q >> 5;                      // 0..3
    float acc = mu_b[(size_t)o * FOUT + g];
#pragma unroll 8
    for (int k = 0; k < FIN; ++k)
      acc = fmaf(xs[bb * XS_STRIDE + k], mwl[k * FOUT + g], acc);
    mus[bb * MU_STRIDE + g] = acc;
  }
  __syncthreads();

  // ---- epilogue: out[m,b,o,l] = mu[b,l] + sum_f sigma[b, l*4+f] * eps[m,b,o,f] ----
  float4* outv = (float4*)out;
  const float* sg = &sig[b0 * SIG_STRIDE];
  const float* mm = &mus[b0 * MU_STRIDE];
#pragma unroll
  for (int i = 0; i < 2; ++i) {
    const int   m = m0 + 8 * i;
    const float4 e = i ? e1 : e0;
    float4 r;
    r.x = mm[0] + sg[ 0] * e.x + sg[ 1] * e.y + sg[ 2] * e.z + sg[ 3] * e.w;
    r.y = mm[1] + sg[ 4] * e.x + sg[ 5] * e.y + sg[ 6] * e.z + sg[ 7] * e.w;
    r.z = mm[2] + sg[ 8] * e.x + sg[ 9] * e.y + sg[10] * e.z + sg[11] * e.w;
    r.w = mm[3] + sg[12] * e.x + sg[13] * e.y + sg[14] * e.z + sg[15] * e.w;
    outv[((size_t)m * B_ + b0) * O_ + o] = r;
  }
}

extern "C" void kernel_launch(void* const* d_in, const int* in_sizes, int n_in,
                              void* d_out, int out_size, void* d_ws, size_t ws_size,
                              hipStream_t stream) {
  (void)in_sizes; (void)n_in; (void)d_ws; (void)ws_size; (void)out_size;
  const float* x       = (const float*)d_in[0];
  const float* mu_w    = (const float*)d_in[1];
  const float* mu_b    = (const float*)d_in[2];
  const float* sigma_w = (const float*)d_in[3];
  const float* sigma_b = (const float*)d_in[4];
  const float* eps     = (const float*)d_in[5];
  float* out = (float*)d_out;

  mgr_fused<<<O_, 256, 0, stream>>>(x, mu_w, mu_b, sigma_w, sigma_b, eps, out);
}